// Shallow_Water_Model_11407433138758
// MI455X (gfx1250) — compile-verified
//
#include <hip/hip_runtime.h>
#include <stdint.h>

// Shallow-water RHS, 4096x4096 periodic, fp32. Memory-bound: 402 MB -> ~17us
// floor @ 23.3 TB/s; WMMA irrelevant (1.5 GFLOP). CDNA5 data movement:
//   - TDM tensor_load_to_lds stages each block's 3 field tiles (with LDS
//     row padding done by the TDM pad feature), waited via s_wait_tensorcnt
//   - per-lane global_load_async_to_lds for periodic-wrap rows + halo columns
//   - nontemporal b128 stores for the write-once output

#define NN 4096
#define TX 128                 // tile width
#define TY 8                   // tile height
#define LROWS (TY + 2)
#define LPITCH 136             // 544 B pitch (16B aligned); 128 payload + 8 pad
#define LOFF 4                 // interior starts 16B-aligned

typedef float v4f __attribute__((ext_vector_type(4)));
typedef int   g4i  __attribute__((vector_size(16)));
typedef unsigned int u32x4 __attribute__((vector_size(16)));
typedef int   i32x4 __attribute__((vector_size(16)));
typedef int   i32x8 __attribute__((vector_size(32)));

typedef __attribute__((address_space(3))) g4i*  lds_v4i_ptr;
typedef __attribute__((address_space(3))) int*  lds_i32_ptr;
typedef __attribute__((address_space(3))) void* lds_void_ptr;

#if __has_builtin(__builtin_amdgcn_tensor_load_to_lds)
#define HAVE_TDM 1
#else
#define HAVE_TDM 0
#endif

__device__ __forceinline__ void async_copy_b128(const float* g, float* l) {
#if __has_builtin(__builtin_amdgcn_global_load_async_to_lds_b128)
  __builtin_amdgcn_global_load_async_to_lds_b128((g4i*)g, (lds_v4i_ptr)l, 0, 0);
#else
  *(v4f*)l = *(const v4f*)g;
#endif
}

__device__ __forceinline__ void async_copy_b32(const float* g, float* l) {
#if __has_builtin(__builtin_amdgcn_global_load_async_to_lds_b32)
  __builtin_amdgcn_global_load_async_to_lds_b32((int*)g, (lds_i32_ptr)l, 0, 0);
#else
  *l = *g;
#endif
}

__device__ __forceinline__ void async_wait_all() {
#if __has_builtin(__builtin_amdgcn_s_wait_asynccnt)
  __builtin_amdgcn_s_wait_asynccnt(0);
#else
  asm volatile("s_wait_asynccnt 0x0" ::: "memory");
#endif
}

#if HAVE_TDM
// One TDM descriptor: rows x 128 fp32 tile, global row stride 4096 elements,
// LDS destination padded 8 DWORDs every 128 DWORDs -> lands on LPITCH rows.
__device__ __forceinline__ void tdm_load_tile(const float* gsrc, float* ldst, int rows) {
  const uint64_t ga = (uint64_t)(uintptr_t)gsrc;
  const uint32_t la = (uint32_t)(uintptr_t)(lds_void_ptr)ldst;

  u32x4 g0;
  g0[0] = 1u;                                                  // count=1 (valid), user
  g0[1] = la;                                                  // lds_addr (bytes)
  g0[2] = (uint32_t)ga;                                        // global_addr[31:0]
  g0[3] = (uint32_t)((ga >> 32) & 0x01FFFFFFu) | (2u << 30);   // addr[56:32] | type=2

  i32x8 g1;
  g1[0] = (int)((2u << 16)        // data_size = 4 bytes
              | (1u << 20)        // pad_enable
              | (6u << 22)        // pad_interval: every 128 DWORDs
              | (7u << 25));      // pad_amount: 8 DWORDs
  g1[1] = (int)((uint32_t)TX << 16);     // tensor_dim0[15:0]=128 (barrier addr=0)
  g1[2] = (int)((uint32_t)rows << 16);   // tensor_dim0[31:16]=0 | tensor_dim1[15:0]=rows
  g1[3] = (int)((uint32_t)TX << 16);     // tensor_dim1[31:16]=0 | tile_dim0=128
  g1[4] = rows;                          // tile_dim1=rows | tile_dim2=0
  g1[5] = NN;                            // tensor_dim0_stride[31:0]=4096 elements
  g1[6] = 0;                             // stride hi | tensor_dim1_stride lo
  g1[7] = 0;

  i32x4 gz = {0, 0, 0, 0};               // 2D tensor: groups 2/3 unused
#if defined(__clang_major__) && (__clang_major__ >= 23)
  i32x8 gz8 = {0, 0, 0, 0, 0, 0, 0, 0};
  __builtin_amdgcn_tensor_load_to_lds(g0, g1, gz, gz, gz8, 0);
#else
  __builtin_amdgcn_tensor_load_to_lds(g0, g1, gz, gz, 0);
#endif
}
#endif

__global__ __launch_bounds__(256)
void swe_rhs_kernel(const float* __restrict__ state, float* __restrict__ out) {
  __shared__ float lds[3][LROWS][LPITCH];   // 16320 B

  const int tx = threadIdx.x;               // 0..31
  const int ty = threadIdx.y;               // 0..7 (== wave id)
  const int tid = ty * 32 + tx;
  const int x0 = blockIdx.x * TX;
  const int y0 = blockIdx.y * TY;
  const int lastby = (int)gridDim.y - 1;
  const int rlo = ((int)blockIdx.y == 0) ? 1 : 0;            // skip wrap row(s) for TDM
  const int rhi = ((int)blockIdx.y == lastby) ? (LROWS - 1) : LROWS;

#if HAVE_TDM
  // ---- bulk staging: 3 TDM descriptors issued by wave 0 only ----
  if (ty == 0) {
    const int gRow0 = y0 - 1 + rlo;        // always in [0, NN-1] by construction
    #pragma unroll
    for (int f = 0; f < 3; ++f)
      tdm_load_tile(state + (size_t)f * NN * NN + (size_t)gRow0 * NN + x0,
                    &lds[f][rlo][LOFF], rhi - rlo);
    __builtin_amdgcn_s_wait_tensorcnt(0);  // wave-uniform branch: legal
  }
  // ---- periodic-wrap rows (edge blocks only): per-lane async b128 ----
  if (rlo == 1 && tid < 96) {              // blockIdx.y==0 : row 0 <- gy=NN-1
    const int f = tid >> 5, c4 = tid & 31;
    async_copy_b128(state + (size_t)f * NN * NN + (size_t)(NN - 1) * NN + (x0 + c4 * 4),
                    &lds[f][0][LOFF + c4 * 4]);
  }
  if (rhi == LROWS - 1 && tid < 96) {      // blockIdx.y==last : row 9 <- gy=0
    const int f = tid >> 5, c4 = tid & 31;
    async_copy_b128(state + (size_t)f * NN * NN + (x0 + c4 * 4),
                    &lds[f][LROWS - 1][LOFF + c4 * 4]);
  }
#else
  // ---- fallback: per-lane async staging of all rows (shift/mask indexing) ----
  #pragma unroll
  for (int f = 0; f < 3; ++f) {
    {
      const int r = tid >> 5, c4 = tid & 31;                    // chunks 0..255
      const int gy = (y0 - 1 + r) & (NN - 1);
      async_copy_b128(state + (size_t)f * NN * NN + (size_t)gy * NN + (x0 + c4 * 4),
                      &lds[f][r][LOFF + c4 * 4]);
    }
    if (tid < 64) {                                             // chunks 256..319
      const int i = tid + 256;
      const int r = i >> 5, c4 = i & 31;
      const int gy = (y0 - 1 + r) & (NN - 1);
      async_copy_b128(state + (size_t)f * NN * NN + (size_t)gy * NN + (x0 + c4 * 4),
                      &lds[f][r][LOFF + c4 * 4]);
    }
  }
#endif
  // ---- x-halo columns: 3 fields * 10 rows * 2 sides = 60 per-lane async b32 ----
  if (tid < 3 * LROWS * 2) {
    int q = tid;                           // q = f*20 + r*2 + side
    const int side = q & 1;  q >>= 1;      // q = f*10 + r
    const int f = (q * 13) >> 7;           // q/10 for q<30 via mul-shift (exact)
    const int r = q - f * 10;
    const int gy = (y0 - 1 + r) & (NN - 1);
    const int gx = side ? ((x0 + TX) & (NN - 1)) : ((x0 - 1) & (NN - 1));
    async_copy_b32(state + (size_t)f * NN * NN + (size_t)gy * NN + gx,
                   &lds[f][r][side ? (LOFF + TX) : (LOFF - 1)]);
  }
  async_wait_all();        // this wave's async LDS writes complete
  __syncthreads();         // TDM + all waves' async tiles visible

  // ---- compute from LDS; each thread: 4 consecutive x-points, 3 outputs ----
  const float inv2d = 0.5f * (float)NN;        // 1/(2*DX) = 2048
  const float invd2 = (float)NN * (float)NN;   // 1/DX^2
  const float Hd = 1.0f, Gg = 9.8f, Mu = 1e-4f;

  const int r  = ty + 1;
  const int c0 = LOFF + tx * 4;
  const float* hN = lds[0][r - 1]; const float* hC = lds[0][r]; const float* hS = lds[0][r + 1];
  const float* uN = lds[1][r - 1]; const float* uC = lds[1][r]; const float* uS = lds[1][r + 1];
  const float* vN = lds[2][r - 1]; const float* vC = lds[2][r]; const float* vS = lds[2][r + 1];

  v4f dh, du, dv;
#pragma unroll
  for (int j = 0; j < 4; ++j) {
    const int c = c0 + j;
    const float he = hC[c + 1], hw = hC[c - 1], hn = hN[c], hs = hS[c];
    const float uc = uC[c], ue = uC[c + 1], uw = uC[c - 1], un = uN[c], us = uS[c];
    const float vc = vC[c], ve = vC[c + 1], vw = vC[c - 1], vn = vN[c], vs = vS[c];
    // dh = -d1x((H+h)u) - d1y((H+h)v)     (roll(-1) == index+1)
    dh[j] = -((Hd + he) * ue - (Hd + hw) * uw) * inv2d
            -((Hd + hs) * vs - (Hd + hn) * vn) * inv2d;
    // du = MU*(d2x(u)+d2y(u)) - G*d1x(h)
    du[j] = Mu * ((ue - 2.0f * uc + uw) + (us - 2.0f * uc + un)) * invd2
            - Gg * (he - hw) * inv2d;
    dv[j] = Mu * ((ve - 2.0f * vc + vw) + (vs - 2.0f * vc + vn)) * invd2
            - Gg * (hs - hn) * inv2d;
  }

  // write-once outputs: nontemporal b128 stores
  const size_t o = (size_t)(y0 + ty) * NN + (size_t)(x0 + tx * 4);
  __builtin_nontemporal_store(dh, (v4f*)(out + o));
  __builtin_nontemporal_store(du, (v4f*)(out + (size_t)NN * NN + o));
  __builtin_nontemporal_store(dv, (v4f*)(out + 2 * (size_t)NN * NN + o));
}

extern "C" void kernel_launch(void* const* d_in, const int* in_sizes, int n_in,
                              void* d_out, int out_size, void* d_ws, size_t ws_size,
                              hipStream_t stream) {
  // d_in[0] = t (unused scalar), d_in[1] = state[3][4096][4096] fp32
  const float* state = (const float*)d_in[1];
  float* out = (float*)d_out;
  (void)in_sizes; (void)n_in; (void)out_size; (void)d_ws; (void)ws_size;

  dim3 grid(NN / TX, NN / TY);   // 32 x 512 blocks
  dim3 block(32, TY);            // 256 threads = 8 waves (wave32)
  swe_rhs_kernel<<<grid, block, 0, stream>>>(state, out);
}